// deltaPlusModule_47021301957201
// MI455X (gfx1250) — compile-verified
//
#include <hip/hip_runtime.h>
#include <math.h>

// ---------------- constants (folded from the reference) ----------------
// h = 0.05, C = 7/(pi h^2) = 2800/pi
#define KC        891.2676813146104f      // C
#define NEG20CH  -356507.07252584404f     // -20 * C / h
#define INV_K0N   5.3333333333333333f     // 1 / ((1-0.5)^4 * (1+4*0.5)) = 1/0.1875
#define RCOEF     0.2f                    // R
#define SCALE_PRE -0.015f                 // -CFL * 4h^2 = -1.5 * 0.01
#define INV_C0    0.041218424918447214f   // 1 / (10*sqrt(2*9.81*0.3))

#define TPB 256

// ---------------- CDNA5 async-to-LDS plumbing (guarded) ----------------
#if defined(__AMDGCN__) && __has_builtin(__builtin_amdgcn_global_load_async_to_lds_b32) && __has_builtin(__builtin_amdgcn_global_load_async_to_lds_b64)
#define HAVE_ASYNC_LDS 1

typedef int v2i_t __attribute__((vector_size(8)));
typedef __attribute__((address_space(1))) int    g_i32;
typedef __attribute__((address_space(3))) int    l_i32;
typedef __attribute__((address_space(1))) v2i_t  g_v2i;
typedef __attribute__((address_space(3))) v2i_t  l_v2i;

__device__ __forceinline__ void async_g2l_b32(const void* g, void* l) {
    __builtin_amdgcn_global_load_async_to_lds_b32((g_i32*)g, (l_i32*)l, 0, 0);
}
__device__ __forceinline__ void async_g2l_b64(const void* g, void* l) {
    __builtin_amdgcn_global_load_async_to_lds_b64((g_v2i*)g, (l_v2i*)l, 0, 0);
}
#if __has_builtin(__builtin_amdgcn_s_wait_asynccnt)
#define WAIT_ASYNCCNT(n) do { __builtin_amdgcn_s_wait_asynccnt(n); asm volatile("" ::: "memory"); } while (0)
#else
#define WAIT_ASYNCCNT(n) asm volatile("s_wait_asynccnt %0" :: "n"(n) : "memory")
#endif
#endif

// ---------------- shared edge math ----------------
// Returns coef such that contribution = coef * (dx, dy); caller applies sign.
__device__ __forceinline__ float edge_coef(float rad, float mass, float scale) {
    float q  = fminf(fmaxf(rad, 0.0f), 1.0f);
    float t  = 1.0f - q;
    float t2 = t * t;
    float t4 = t2 * t2;
    float u  = t4 * fmaf(4.0f, q, 1.0f) * INV_K0N;   // W(q)/k0  (C cancels)
    float u2 = u * u;
    float kT = fmaf(RCOEF, u2 * u2, 1.0f);           // 1 + R*(W/k0)^4
    float dWdr = NEG20CH * q * (t * t2);             // (C/h) * (-20 q (1-q)^3)
    return scale * kT * mass * dWdr;
}

// ---------------- kernels (fluid edge kernel first so it leads the disasm) ----
__global__ void __launch_bounds__(TPB) fluid_edges_kernel(
    const float* __restrict__ rad, const float* __restrict__ dist,
    const int* __restrict__ eI, const int* __restrict__ eJ,
    const float* __restrict__ area, const float* __restrict__ rho,
    const float* __restrict__ wsf, float* __restrict__ out, int E) {
    const float scale = wsf[1];
    const int tid = threadIdx.x;
#ifdef HAVE_ASYNC_LDS
    // Double-buffered async-to-LDS streaming of the 20B/edge payload.
    __shared__ float  s_rad[2][TPB];
    __shared__ float2 s_dst[2][TPB];
    __shared__ int    s_i[2][TPB];
    __shared__ int    s_j[2][TPB];
    const int nTiles = E / TPB;    // E is a multiple of 256
    int tile = blockIdx.x;
    if (tile < nTiles) {
        int e = tile * TPB + tid;
        async_g2l_b32(rad + e,        &s_rad[0][tid]);
        async_g2l_b64(dist + 2 * e,   &s_dst[0][tid]);
        async_g2l_b32(eI + e,         &s_i[0][tid]);
        async_g2l_b32(eJ + e,         &s_j[0][tid]);
    }
    int buf = 0;
    for (; tile < nTiles; tile += gridDim.x) {
        int next = tile + gridDim.x;
        if (next < nTiles) {
            int e2 = next * TPB + tid;
            async_g2l_b32(rad + e2,      &s_rad[buf ^ 1][tid]);
            async_g2l_b64(dist + 2 * e2, &s_dst[buf ^ 1][tid]);
            async_g2l_b32(eI + e2,       &s_i[buf ^ 1][tid]);
            async_g2l_b32(eJ + e2,       &s_j[buf ^ 1][tid]);
            WAIT_ASYNCCNT(4);   // current tile's 4 loads done (in-order completion)
        } else {
            WAIT_ASYNCCNT(0);
        }
        float  r   = s_rad[buf][tid];
        float2 dxy = s_dst[buf][tid];
        int ii = s_i[buf][tid];
        int jj = s_j[buf][tid];
        float mass = area[jj] / (rho[ii] + rho[jj]);
        float coef = edge_coef(r, mass, scale);
        atomicAdd(&out[2 * ii + 0], coef * dxy.x);   // global_atomic_add_f32 (no return)
        atomicAdd(&out[2 * ii + 1], coef * dxy.y);
        buf ^= 1;
    }
#else
    int stride = gridDim.x * blockDim.x;
    for (int e = blockIdx.x * blockDim.x + tid; e < E; e += stride) {
        float  r   = rad[e];
        float2 dxy = ((const float2*)dist)[e];
        int ii = eI[e], jj = eJ[e];
        float mass = area[jj] / (rho[ii] + rho[jj]);
        float coef = edge_coef(r, mass, scale);
        atomicAdd(&out[2 * ii + 0], coef * dxy.x);
        atomicAdd(&out[2 * ii + 1], coef * dxy.y);
    }
#endif
}

__global__ void __launch_bounds__(TPB) bdy_edges_kernel(
    const float* __restrict__ rad, const float* __restrict__ dist,
    const int* __restrict__ ebb, const int* __restrict__ ebf,
    const float* __restrict__ vol, const float* __restrict__ rhoB,
    const float* __restrict__ rhoF, const float* __restrict__ wsf,
    float* __restrict__ out, int EB) {
    const float scale = wsf[1];
    int e = blockIdx.x * blockDim.x + threadIdx.x;
    if (e >= EB) return;
    float  r   = rad[e];
    float2 dxy = ((const float2*)dist)[e];
    int b = ebb[e], f = ebf[e];
    float mass = vol[b] / (rhoF[f] + rhoB[b]);
    float coef = -edge_coef(r, mass, scale);   // gradient uses -distances
    atomicAdd(&out[2 * f + 0], coef * dxy.x);
    atomicAdd(&out[2 * f + 1], coef * dxy.y);
}

__global__ void __launch_bounds__(TPB) velmax_kernel(const float* __restrict__ vel,
                                                     unsigned* __restrict__ wsu, int N) {
    int stride = gridDim.x * blockDim.x;
    float m = 0.0f;
    for (int idx = blockIdx.x * blockDim.x + threadIdx.x; idx < N; idx += stride) {
        __builtin_prefetch(&vel[2 * (idx + 4 * stride)], 0, 0);   // global_prefetch_b8
        float vx = vel[2 * idx + 0];
        float vy = vel[2 * idx + 1];
        m = fmaxf(m, fmaf(vx, vx, vy * vy));                      // max of |v|^2
    }
    // wave32 butterfly reduction
    for (int off = 16; off > 0; off >>= 1)
        m = fmaxf(m, __shfl_xor(m, off, 32));
    __shared__ float sm[TPB / 32];
    int lane = threadIdx.x & 31, wave = threadIdx.x >> 5;
    if (lane == 0) sm[wave] = m;
    __syncthreads();
    if (threadIdx.x == 0) {
        float mm = sm[0];
        for (int w = 1; w < (TPB >> 5); ++w) mm = fmaxf(mm, sm[w]);
        atomicMax(wsu, __float_as_uint(mm));   // valid ordering: mm >= 0
    }
}

__global__ void __launch_bounds__(TPB) init_kernel(float* __restrict__ out,
                                                   unsigned* __restrict__ wsu, int n2) {
    int idx = blockIdx.x * blockDim.x + threadIdx.x;
    if (idx < n2) out[idx] = 0.0f;
    if (idx == 0) wsu[0] = 0u;
}

__global__ void scale_kernel(float* __restrict__ wsf) {
    unsigned bits = ((unsigned*)wsf)[0];
    float maxnorm = sqrtf(__uint_as_float(bits));
    wsf[1] = SCALE_PRE * (maxnorm * INV_C0);   // -CFL * Ma * 4h^2
}

// ---------------- launch ----------------
extern "C" void kernel_launch(void* const* d_in, const int* in_sizes, int n_in,
                              void* d_out, int out_size, void* d_ws, size_t ws_size,
                              hipStream_t stream) {
    const float* vel   = (const float*)d_in[0];
    const float* area  = (const float*)d_in[1];
    const float* rho   = (const float*)d_in[2];
    const float* radF  = (const float*)d_in[3];
    const float* distF = (const float*)d_in[4];
    const float* volB  = (const float*)d_in[5];
    const float* rhoB  = (const float*)d_in[6];
    const float* radB  = (const float*)d_in[7];
    const float* distB = (const float*)d_in[8];
    const int*   eI    = (const int*)d_in[9];
    const int*   eJ    = (const int*)d_in[10];
    const int*   ebb   = (const int*)d_in[11];
    const int*   ebf   = (const int*)d_in[12];
    float*    out = (float*)d_out;
    unsigned* wsu = (unsigned*)d_ws;
    float*    wsf = (float*)d_ws;

    int N  = in_sizes[1];   // fluidArea count
    int E  = in_sizes[3];   // fluid edges
    int EB = in_sizes[7];   // boundary edges
    int n2 = out_size;      // 2*N

    init_kernel<<<(n2 + TPB - 1) / TPB, TPB, 0, stream>>>(out, wsu, n2);
    velmax_kernel<<<512, TPB, 0, stream>>>(vel, wsu, N);
    scale_kernel<<<1, 1, 0, stream>>>(wsf);
    fluid_edges_kernel<<<2048, TPB, 0, stream>>>(radF, distF, eI, eJ, area, rho,
                                                 wsf, out, E);
    bdy_edges_kernel<<<(EB + TPB - 1) / TPB, TPB, 0, stream>>>(radB, distB, ebb, ebf,
                                                               volB, rhoB, rho, wsf,
                                                               out, EB);
}